// AttentionHeadCheb_62577673503657
// MI455X (gfx1250) — compile-verified
//
#include <hip/hip_runtime.h>
#include <math.h>

#define NN   4096
#define DIN  256
#define DOUT 256
#define KK   3
#define CAP  2048

typedef __attribute__((ext_vector_type(2))) float v2f;
typedef __attribute__((ext_vector_type(8))) float v8f;

// -------------------------------------------------------------------------
// Kernel 1: wx[k] = x @ W_t[k]   (fp32, V_WMMA_F32_16X16X4_F32)
// One wave per 16x16 output tile; K-loop in steps of 4.
// A (16x4): lanes 0-15 hold M=0..15 with K={0,1}; lanes 16-31 K={2,3}.
// B (4x16): VGPR0 = rows K={0|2} across lanes (cols), VGPR1 = rows K={1|3}.
// D (16x16): VGPR v -> row M=v (lanes 0-15), M=v+8 (lanes 16-31).
// -------------------------------------------------------------------------
__global__ __launch_bounds__(128) void gemm_xwt(const float* __restrict__ x,
                                                const float* __restrict__ Wt,
                                                float* __restrict__ wx) {
  const int lane  = threadIdx.x & 31;
  const int wid   = threadIdx.x >> 5;
  const int tileN = blockIdx.x * 4 + wid;   // 0..15   (DOUT/16)
  const int tileM = blockIdx.y;             // 0..255  (N/16)
  const int k     = blockIdx.z;             // 0..2
  const int hi = lane >> 4;                 // half-wave select
  const int lo = lane & 15;

  const float* A = x + (size_t)(tileM * 16 + lo) * DIN;
  const float* B = Wt + (size_t)k * DIN * DOUT + tileN * 16 + lo;

  v8f c = {};
  for (int kk = 0; kk < DIN; kk += 4) {
    v2f a = *(const v2f*)(A + kk + 2 * hi);       // K = 2*hi, 2*hi+1
    v2f b;
    b.x = B[(size_t)(kk + 2 * hi + 0) * DOUT];
    b.y = B[(size_t)(kk + 2 * hi + 1) * DOUT];
    c = __builtin_amdgcn_wmma_f32_16x16x4_f32(false, a, false, b,
                                              (short)0, c, false, false);
  }

  float* D = wx + (size_t)k * NN * DOUT
                + (size_t)(tileM * 16 + 8 * hi) * DOUT + tileN * 16 + lo;
#pragma unroll
  for (int v = 0; v < 8; ++v)
    D[(size_t)v * DOUT] = c[v];
}

// -------------------------------------------------------------------------
// Kernel 2: al[k][n][j] = wx[k][n][:] . W_l[k][j][:]   (and same for ar)
// One wave per (k,n); lane-strided dot products + shfl_xor reduction.
// -------------------------------------------------------------------------
__global__ __launch_bounds__(128) void proj_alar(const float* __restrict__ wx,
                                                 const float* __restrict__ Wl,
                                                 const float* __restrict__ Wr,
                                                 float* __restrict__ al,
                                                 float* __restrict__ ar) {
  const int lane = threadIdx.x & 31;
  const int wid  = threadIdx.x >> 5;
  const int kn   = blockIdx.x * 4 + wid;       // k*4096 + n
  const int k    = kn >> 12;

  const float* row = wx + (size_t)kn * DOUT;
  float pl[4] = {0.f, 0.f, 0.f, 0.f};
  float pr[4] = {0.f, 0.f, 0.f, 0.f};
  for (int d = lane; d < DOUT; d += 32) {
    float v = row[d];
#pragma unroll
    for (int j = 0; j < 4; ++j) {
      pl[j] += v * Wl[((k * 4) + j) * DOUT + d];
      pr[j] += v * Wr[((k * 4) + j) * DOUT + d];
    }
  }
#pragma unroll
  for (int j = 0; j < 4; ++j) {
#pragma unroll
    for (int off = 16; off > 0; off >>= 1) {
      pl[j] += __shfl_xor(pl[j], off, 32);
      pr[j] += __shfl_xor(pr[j], off, 32);
    }
  }
  if (lane == 0) {
#pragma unroll
    for (int j = 0; j < 4; ++j) {
      al[(size_t)kn * 4 + j] = pl[j];
      ar[(size_t)kn * 4 + j] = pr[j];
    }
  }
}

// -------------------------------------------------------------------------
// Kernel 3: per-node candidate scan (single pass over all 6 mask rows),
// masked softmax per k, sparse aggregation of wx rows, fused ELU.
// One 256-thread block per node; thread == output channel in aggregation.
// -------------------------------------------------------------------------
__global__ __launch_bounds__(256) void attn_rows(const int* __restrict__ sup,
                                                 const int* __restrict__ asup,
                                                 const float* __restrict__ wx,
                                                 const float* __restrict__ al,
                                                 const float* __restrict__ ar,
                                                 float* __restrict__ out) {
  __shared__ int   tmpcode[256 * 16];  // per-thread candidate staging (16 KB)
  __shared__ int   cand[CAP];          // compacted codes (bits<<12)|m (8 KB)
  __shared__ float sval[CAP];          // scores / probs (8 KB)
  __shared__ int   scan[256];
  __shared__ float red[256];

  const int n   = blockIdx.x;
  const int tid = threadIdx.x;

  const size_t rowoff = (size_t)n * NN;
  const int* a0 = asup + 0 * (size_t)NN * NN + rowoff;
  const int* a1 = asup + 1 * (size_t)NN * NN + rowoff;
  const int* a2 = asup + 2 * (size_t)NN * NN + rowoff;
  const int* s0 = sup  + 0 * (size_t)NN * NN + rowoff;
  const int* s1 = sup  + 1 * (size_t)NN * NN + rowoff;
  const int* s2 = sup  + 2 * (size_t)NN * NN + rowoff;

  // ---- single pass over masks: thread owns columns [tid*16, tid*16+16) ----
  const int mbase = tid * 16;
  int mc = 0;
#pragma unroll
  for (int i = 0; i < 4; ++i) {
    const int m0 = mbase + i * 4;
    int4 va0 = *(const int4*)(a0 + m0);
    int4 va1 = *(const int4*)(a1 + m0);
    int4 va2 = *(const int4*)(a2 + m0);
    int4 vs0 = *(const int4*)(s0 + m0);
    int4 vs1 = *(const int4*)(s1 + m0);
    int4 vs2 = *(const int4*)(s2 + m0);
#define PUSH(comp, off)                                                       \
    {                                                                         \
      int b = (va0.comp ? 1 : 0) | (va1.comp ? 2 : 0) | (va2.comp ? 4 : 0) |  \
              (vs0.comp ? 8 : 0) | (vs1.comp ? 16 : 0) | (vs2.comp ? 32 : 0); \
      if (b) tmpcode[mbase + mc++] = (b << 12) | (m0 + (off));                \
    }
    PUSH(x, 0) PUSH(y, 1) PUSH(z, 2) PUSH(w, 3)
#undef PUSH
  }

  // ---- deterministic compaction via Hillis-Steele inclusive scan ----
  scan[tid] = mc;
  __syncthreads();
  for (int off = 1; off < 256; off <<= 1) {
    int v = (tid >= off) ? scan[tid - off] : 0;
    __syncthreads();
    scan[tid] += v;
    __syncthreads();
  }
  const int wbase = scan[tid] - mc;
  const int count = min(scan[255], CAP);
  for (int j = 0; j < mc; ++j) {
    int idx = wbase + j;
    if (idx < CAP) cand[idx] = tmpcode[mbase + j];
  }
  __syncthreads();

  // ---- per-k masked softmax + aggregation ----
  float acc_total = 0.f;
  const int c = tid;  // output channel
  for (int k = 0; k < KK; ++k) {
    const float4 aln = *(const float4*)(al + ((size_t)k * NN + n) * 4);
    const float4* ark = (const float4*)(ar + (size_t)k * NN * 4);

    for (int e = tid; e < count; e += 256) {
      const int code = cand[e];
      const int m = code & 4095;
      const int b = code >> 12;
      const float4 arm = ark[m];
      float s = 0.f;
      if (b & 1) s += aln.x + arm.x;
      if (b & 2) s += aln.y + arm.y;
      if (b & 4) s += aln.z + arm.z;
      if (b & (8 << k)) s += aln.w + arm.w;   // supports[k] term
      sval[e] = s;
    }
    __syncthreads();

    // block max over valid entries (score != 0, matching reference)
    float mx = -3.0e38f;
    for (int e = tid; e < count; e += 256) {
      float s = sval[e];
      if (s != 0.f) mx = fmaxf(mx, s);
    }
    red[tid] = mx;
    __syncthreads();
    for (int off = 128; off > 0; off >>= 1) {
      if (tid < off) red[tid] = fmaxf(red[tid], red[tid + off]);
      __syncthreads();
    }
    mx = red[0];
    __syncthreads();

    // exp + block sum
    float psum = 0.f;
    for (int e = tid; e < count; e += 256) {
      float s = sval[e];
      float p = (s != 0.f) ? __expf(s - mx) : 0.f;
      sval[e] = p;
      psum += p;
    }
    red[tid] = psum;
    __syncthreads();
    for (int off = 128; off > 0; off >>= 1) {
      if (tid < off) red[tid] += red[tid + off];
      __syncthreads();
    }
    const float Z = red[0];
    __syncthreads();

    if (mx > -2.9e38f && Z > 0.f) {   // at least one valid edge
      const float inv = 1.f / Z;
      const float* wxk = wx + (size_t)k * NN * DOUT;
      float acc = 0.f;
      for (int e = 0; e < count; ++e) {     // uniform loop -> coalesced rows
        float p = sval[e];
        if (p != 0.f) {
          int m = cand[e] & 4095;
          acc += p * wxk[(size_t)m * DOUT + c];
        }
      }
      acc_total += acc * inv;
    }
    __syncthreads();   // before next k overwrites sval
  }

  const float v = acc_total;
  out[(size_t)n * DOUT + c] = (v > 0.f) ? v : expm1f(v);   // ELU(alpha=1)
}

// -------------------------------------------------------------------------
extern "C" void kernel_launch(void* const* d_in, const int* in_sizes, int n_in,
                              void* d_out, int out_size, void* d_ws, size_t ws_size,
                              hipStream_t stream) {
  const float* x    = (const float*)d_in[0];
  const int*   sup  = (const int*)d_in[1];   // supports [K,N,N]
  const int*   asup = (const int*)d_in[2];   // atten_supports [R,N,N]
  const float* Wt   = (const float*)d_in[3]; // [K, DIN, DOUT]
  const float* Wl   = (const float*)d_in[4]; // [K, 4, DOUT]
  const float* Wr   = (const float*)d_in[5]; // [K, 4, DOUT]
  float* out = (float*)d_out;

  float* wx = (float*)d_ws;                          // [K][N][DOUT] 12.6 MB
  float* al = wx + (size_t)KK * NN * DOUT;           // [K][N][4]
  float* ar = al + (size_t)KK * NN * 4;              // [K][N][4]

  dim3 g1(DOUT / 16 / 4, NN / 16, KK);
  gemm_xwt<<<g1, 128, 0, stream>>>(x, Wt, wx);
  proj_alar<<<(KK * NN) / 4, 128, 0, stream>>>(wx, Wl, Wr, al, ar);
  attn_rows<<<NN, 256, 0, stream>>>(sup, asup, wx, al, ar, out);
}